// GNNLayer_37160057045128
// MI455X (gfx1250) — compile-verified
//
#include <hip/hip_runtime.h>
#include <hip/hip_bf16.h>
#include <math.h>

// ---------------------------------------------------------------------------
// GatedGCN layer for MI455X (gfx1250), wave32, WMMA f32 16x16x4 +
// async global->LDS staging (ASYNCcnt) for the edge GEMM.
// B=4, V=256, H=128.
// ---------------------------------------------------------------------------

typedef __attribute__((ext_vector_type(2))) float v2f;
typedef __attribute__((ext_vector_type(8))) float v8f;
typedef __attribute__((ext_vector_type(4))) int v4i_;

#define B_ 4
#define V_ 256
#define H_ 128
#define NV (B_ * V_)          // 1024 node rows
#define NE (B_ * V_ * V_)     // 262144 edge rows
#define NODE_F (NV * H_)      // 131072 floats per node tensor
#define EDGE_F (NE * H_)      // 33554432 floats

// workspace float offsets
#define WS_UH1   0
#define WS_UH2   (1 * NODE_F)
#define WS_VH1   (2 * NODE_F)
#define WS_VH2   (3 * NODE_F)
#define WS_AH1   (4 * NODE_F)
#define WS_AH2   (5 * NODE_F)
#define WS_H1PRE (6 * NODE_F)
#define WS_H2PRE (7 * NODE_F)
#define WS_PSUM  (8 * NODE_F)       // [1024][128] partial BN sums for e
#define WS_PSQ   (9 * NODE_F)       // [1024][128]
#define WS_ESC   (10 * NODE_F)      // [128] scale
#define WS_ESH   (10 * NODE_F + H_) // [128] shift

// ---- CDNA5 async global->LDS (ASYNCcnt) --------------------------------
#if __has_builtin(__builtin_amdgcn_global_load_async_to_lds_b128) && \
    __has_builtin(__builtin_amdgcn_s_wait_asynccnt)
#define HAVE_ASYNC_LDS 1
typedef __attribute__((address_space(1))) v4i_ GV4;   // global int32x4
typedef __attribute__((address_space(3))) v4i_ LV4;   // LDS int32x4
// builtin signature (from clang diagnostic): (v4i AS1*, v4i AS3*, Ii, Ii)
#define ASYNC_CP16(gp, lp) \
  __builtin_amdgcn_global_load_async_to_lds_b128((GV4*)((float*)(gp)), \
                                                 (LV4*)(lp), 0, 0)
#define WAIT_ASYNC(n) __builtin_amdgcn_s_wait_asynccnt(n)
#else
#define HAVE_ASYNC_LDS 0
#endif

static __device__ __forceinline__ v8f wmma4(v2f a, v2f b, v8f c) {
  // V_WMMA_F32_16X16X4_F32: D = A(16x4,f32) x B(4x16,f32) + C(16x16,f32)
  return __builtin_amdgcn_wmma_f32_16x16x4_f32(false, a, false, b, (short)0, c,
                                               false, false);
}

static __device__ __forceinline__ float sigmoidf_(float x) {
  return 1.0f / (1.0f + __expf(-x));
}

// ---------------------------------------------------------------------------
// K1: six node linears: out[z] = X_z @ W_z^T + b_z  (1024x128 @ 128x128)
// grid (64 row-tiles, 8 h-tiles, 6 weights), 1 wave per block.
// Uniform ternary pointer selection keeps W/bias bases in SGPRs (no flat).
// ---------------------------------------------------------------------------
__global__ __launch_bounds__(32) void k_lin6(
    const float* __restrict__ x1, const float* __restrict__ x2,
    const float* __restrict__ W0, const float* __restrict__ W1,
    const float* __restrict__ W2, const float* __restrict__ W3,
    const float* __restrict__ W4, const float* __restrict__ W5,
    const float* __restrict__ c0, const float* __restrict__ c1,
    const float* __restrict__ c2, const float* __restrict__ c3,
    const float* __restrict__ c4, const float* __restrict__ c5,
    float* __restrict__ ws) {
  const int lane = threadIdx.x;
  const int l15 = lane & 15;
  const int hi  = lane >> 4;
  const int rt = blockIdx.x;   // row tile 0..63
  const int ht = blockIdx.y;   // h tile   0..7
  const int w  = blockIdx.z;   // weight   0..5
  const float* Wm = (w == 0) ? W0 : (w == 1) ? W1 : (w == 2) ? W2
                  : (w == 3) ? W3 : (w == 4) ? W4 : W5;
  const float* Bv = (w == 0) ? c0 : (w == 1) ? c1 : (w == 2) ? c2
                  : (w == 3) ? c3 : (w == 4) ? c4 : c5;
  const float* X = (w & 1) ? x2 : x1;   // even -> h1, odd -> h2
  const int hcol = ht * 16 + l15;
  const float* arow = X + (rt * 16 + l15) * H_;  // A: M = lane&15
  const float* brow = Wm + hcol * H_;            // B[k][n] = W[n][k]
  v8f c = {};
#pragma unroll
  for (int kk = 0; kk < H_; kk += 4) {
    v2f a = *(const v2f*)(arow + kk + 2 * hi);   // A: vgpr0=K0/K2, vgpr1=K1/K3
    v2f b = *(const v2f*)(brow + kk + 2 * hi);
    c = wmma4(a, b, c);
  }
  const float bv = Bv[hcol];
  float* out = ws + w * NODE_F;
#pragma unroll
  for (int r = 0; r < 8; ++r) {
    int m = r + 8 * hi;                          // C: vgpr r -> row r / r+8
    out[(rt * 16 + m) * H_ + hcol] = c[r] + bv;
  }
}

// ---------------------------------------------------------------------------
// K2: e_new = e @ W_C^T + b_C + Ah1[b,j,h] + Ah2[b,i,h]
// One wave per ((b,i) pair, 16-h-column tile); loops 16 j-tiles with the
// W_C tile held in 64 VGPRs. A tiles staged through LDS with double-buffered
// async b128 copies (fully coalesced 512B rows); LDS rows padded to 132
// floats so the 16 row-lanes of each ds_load_b64 hit distinct banks.
// grid (1024, 8), 1 wave/block, 512 wmma/wave.
// ---------------------------------------------------------------------------
#define LDS_PITCH 132

__global__ __launch_bounds__(32) void k_edge(
    const float* __restrict__ e, const float* __restrict__ Wc,
    const float* __restrict__ bC, const float* __restrict__ ws,
    float* __restrict__ enew, float* __restrict__ psum,
    float* __restrict__ psq) {
  const int lane = threadIdx.x;
  const int l15 = lane & 15;
  const int hi  = lane >> 4;
  const int rg = blockIdx.x;        // (b,i) pair 0..1023
  const int ht = blockIdx.y;        // 0..7
  const int bIdx = rg >> 8;
  const int hcol = ht * 16 + l15;
  const float* Ah1 = ws + WS_AH1;
  const float* Ah2 = ws + WS_AH2;

  // Preload B tile: W_C[hcol][0..127] as 32 k-steps of 2 floats.
  v2f breg[32];
  const float* brow = Wc + hcol * H_;
#pragma unroll
  for (int kq = 0; kq < 32; ++kq)
    breg[kq] = *(const v2f*)(brow + kq * 4 + 2 * hi);

  const float bCv  = bC[hcol];
  const float ah2v = Ah2[rg * H_ + hcol];   // constant over the row group
  float sacc = 0.f, sqacc = 0.f;

#if HAVE_ASYNC_LDS
  __shared__ float tile[2][16 * LDS_PITCH];
  // issue one 16-row tile: 16 async b128 ops, each = one coalesced 512B row
  auto issue_tile = [&](int jt, int buf) {
    const long rowbase = (long)rg * V_ + jt * 16;
    const float* gsrc = e + rowbase * H_ + lane * 4;
    float* ldst = &tile[buf][lane * 4];
#pragma unroll
    for (int r = 0; r < 16; ++r)
      ASYNC_CP16(gsrc + r * H_, ldst + r * LDS_PITCH);
  };
  issue_tile(0, 0);
#endif

  for (int jt = 0; jt < 16; ++jt) {
    const long rowbase = (long)rg * V_ + jt * 16;      // edge row (b,i,j)
    v8f c = {};
#if HAVE_ASYNC_LDS
    const int buf = jt & 1;
    if (jt < 15) {
      issue_tile(jt + 1, buf ^ 1);
      WAIT_ASYNC(16);                 // oldest 16 (tile jt) complete
    } else {
      WAIT_ASYNC(0);
    }
    __asm__ volatile("" ::: "memory");  // keep ds reads after the wait
    const float* arow = &tile[buf][l15 * LDS_PITCH];
#pragma unroll
    for (int kq = 0; kq < 32; ++kq) {
      v2f a = *(const v2f*)(arow + kq * 4 + 2 * hi);   // ds_load_b64
      c = wmma4(a, breg[kq], c);
    }
#else
    const float* arow = e + (rowbase + l15) * H_;
#pragma unroll
    for (int kq = 0; kq < 32; ++kq) {
      v2f a = *(const v2f*)(arow + kq * 4 + 2 * hi);
      c = wmma4(a, breg[kq], c);
    }
#endif
#pragma unroll
    for (int r = 0; r < 8; ++r) {
      const int m = r + 8 * hi;
      const int j = jt * 16 + m;
      float v = c[r] + bCv + ah2v + Ah1[(bIdx * V_ + j) * H_ + hcol];
      enew[(rowbase + m) * H_ + hcol] = v;
      sacc += v;
      sqacc += v * v;
    }
  }
  // lanes L and L+16 own the same column: fold, then 16 lanes store partials
  sacc  += __shfl_xor(sacc, 16, 32);
  sqacc += __shfl_xor(sqacc, 16, 32);
  if (hi == 0) {
    psum[rg * H_ + hcol] = sacc;
    psq[rg * H_ + hcol]  = sqacc;
  }
}

// ---------------------------------------------------------------------------
// K3: gated aggregation. transposed=0 -> h1 path, 1 -> h2 path.
//   pre[b,i,h] = Uh + Vi[b,i,h]*sum_j sig + sum_j sig*Vj[b,j,h]
// grid 1024 blocks (b,i), 128 threads (h): fully coalesced 512B reads.
// ---------------------------------------------------------------------------
__global__ __launch_bounds__(128) void k_hagg(
    const float* __restrict__ enew, const float* __restrict__ ws,
    float* __restrict__ pre, int transposed) {
  const int h = threadIdx.x;
  const int rg = blockIdx.x;
  const int bIdx = rg >> 8, i = rg & 255;
  const float* Uh = ws + (transposed ? WS_UH2 : WS_UH1);
  const float* Vj = ws + (transposed ? WS_VH2 : WS_VH1);  // per-j multiplier
  const float* Vi = ws + (transposed ? WS_VH1 : WS_VH2);  // per-i constant
  // non-transposed: enew[((b*V+i)*V + j)*H + h]  -> stride H
  // transposed:     enew[((b*V+j)*V + i)*H + h]  -> stride V*H
  const long base   = transposed ? ((long)bIdx * V_ * V_ * H_ + (long)i * H_ + h)
                                 : ((long)rg * V_ * H_ + h);
  const long stride = transposed ? (long)V_ * H_ : (long)H_;
  const float* vjrow = Vj + bIdx * V_ * H_ + h;
  float S = 0.f, T = 0.f;
  for (int j = 0; j < V_; ++j) {
    float s = sigmoidf_(enew[base + j * stride]);
    S += s;
    T += s * vjrow[j * H_];
  }
  pre[rg * H_ + h] = Uh[rg * H_ + h] + Vi[rg * H_ + h] * S + T;
}

// ---------------------------------------------------------------------------
// K4: BN (batch stats) + ReLU + residual for h1 / h2. grid (128 ch, 2).
// ---------------------------------------------------------------------------
__global__ __launch_bounds__(256) void k_bn_h(
    const float* __restrict__ ws, const float* __restrict__ h1in,
    const float* __restrict__ h2in, const float* __restrict__ g1,
    const float* __restrict__ be1, const float* __restrict__ g2,
    const float* __restrict__ be2, float* __restrict__ out) {
  const int ch = blockIdx.x, which = blockIdx.y, t = threadIdx.x;
  const float* pre = ws + (which ? WS_H2PRE : WS_H1PRE);
  const float* hin = which ? h2in : h1in;
  const float* g = which ? g2 : g1;
  const float* be = which ? be2 : be1;
  float* o = out + which * NODE_F;
  float s = 0.f, sq = 0.f;
  for (int r = t; r < NV; r += 256) {
    float v = pre[r * H_ + ch];
    s += v; sq += v * v;
  }
  __shared__ float ss[256], ssq[256];
  ss[t] = s; ssq[t] = sq;
  __syncthreads();
  for (int off = 128; off; off >>= 1) {
    if (t < off) { ss[t] += ss[t + off]; ssq[t] += ssq[t + off]; }
    __syncthreads();
  }
  const float mean = ss[0] * (1.f / NV);
  const float var = ssq[0] * (1.f / NV) - mean * mean;
  const float sc = rsqrtf(var + 1e-5f) * g[ch];
  const float sh = be[ch] - mean * sc;
  for (int r = t; r < NV; r += 256) {
    float v = pre[r * H_ + ch] * sc + sh;
    v = v > 0.f ? v : 0.f;
    o[r * H_ + ch] = v + hin[r * H_ + ch];
  }
}

// ---------------------------------------------------------------------------
// K5: reduce e-BN partials -> per-channel scale/shift. grid 128 blocks.
// ---------------------------------------------------------------------------
__global__ __launch_bounds__(256) void k_bn_e_stats(
    const float* __restrict__ psum, const float* __restrict__ psq,
    const float* __restrict__ ge, const float* __restrict__ bee,
    float* __restrict__ scale, float* __restrict__ shift) {
  const int ch = blockIdx.x, t = threadIdx.x;
  float s = 0.f, sq = 0.f;
  for (int p = t; p < NV; p += 256) {
    s += psum[p * H_ + ch];
    sq += psq[p * H_ + ch];
  }
  __shared__ float ss[256], ssq[256];
  ss[t] = s; ssq[t] = sq;
  __syncthreads();
  for (int off = 128; off; off >>= 1) {
    if (t < off) { ss[t] += ss[t + off]; ssq[t] += ssq[t + off]; }
    __syncthreads();
  }
  if (t == 0) {
    const float mean = ss[0] * (1.f / NE);
    const float var = ssq[0] * (1.f / NE) - mean * mean;
    const float sc = rsqrtf(var + 1e-5f) * ge[ch];
    scale[ch] = sc;
    shift[ch] = bee[ch] - mean * sc;
  }
}

// ---------------------------------------------------------------------------
// K6: e_out = relu(e_new*scale+shift) + e_in, in-place in d_out, float4.
// ---------------------------------------------------------------------------
__global__ __launch_bounds__(256) void k_e_final(
    float* __restrict__ eo, const float* __restrict__ ein,
    const float* __restrict__ scale, const float* __restrict__ shift) {
  const long i4 = (long)blockIdx.x * 256 + threadIdx.x;  // 8388608 total
  const long base = i4 * 4;
  const int hb = (int)(base & (H_ - 1));
  float4 v  = *(const float4*)(eo + base);
  float4 ei = *(const float4*)(ein + base);
  float4 sc = *(const float4*)(scale + hb);
  float4 sh = *(const float4*)(shift + hb);
  float4 o;
  o.x = fmaxf(v.x * sc.x + sh.x, 0.f) + ei.x;
  o.y = fmaxf(v.y * sc.y + sh.y, 0.f) + ei.y;
  o.z = fmaxf(v.z * sc.z + sh.z, 0.f) + ei.z;
  o.w = fmaxf(v.w * sc.w + sh.w, 0.f) + ei.w;
  *(float4*)(eo + base) = o;
}

// ---------------------------------------------------------------------------
extern "C" void kernel_launch(void* const* d_in, const int* in_sizes, int n_in,
                              void* d_out, int out_size, void* d_ws,
                              size_t ws_size, hipStream_t stream) {
  const float* h1 = (const float*)d_in[0];
  const float* h2 = (const float*)d_in[1];
  const float* e  = (const float*)d_in[2];
  // d_in[3] = graph (unused for sum aggregation)
  const float* W_U1 = (const float*)d_in[4];
  const float* b_U1 = (const float*)d_in[5];
  const float* W_U2 = (const float*)d_in[6];
  const float* b_U2 = (const float*)d_in[7];
  const float* W_V1 = (const float*)d_in[8];
  const float* b_V1 = (const float*)d_in[9];
  const float* W_V2 = (const float*)d_in[10];
  const float* b_V2 = (const float*)d_in[11];
  const float* W_A1 = (const float*)d_in[12];
  const float* b_A1 = (const float*)d_in[13];
  const float* W_A2 = (const float*)d_in[14];
  const float* b_A2 = (const float*)d_in[15];
  const float* W_C  = (const float*)d_in[16];
  const float* b_C  = (const float*)d_in[17];
  const float* g_h1 = (const float*)d_in[18];
  const float* be_h1 = (const float*)d_in[19];
  const float* g_h2 = (const float*)d_in[20];
  const float* be_h2 = (const float*)d_in[21];
  const float* g_e  = (const float*)d_in[22];
  const float* be_e = (const float*)d_in[23];

  float* out = (float*)d_out;
  float* ws = (float*)d_ws;
  float* enew = out + 2 * NODE_F;  // stage e_new in d_out's e-region (134MB)

  // K1: six node linears (WMMA f32)
  k_lin6<<<dim3(NV / 16, H_ / 16, 6), 32, 0, stream>>>(
      h1, h2, W_U1, W_U2, W_V1, W_V2, W_A1, W_A2,
      b_U1, b_U2, b_V1, b_V2, b_A1, b_A2, ws);

  // K2: edge GEMM + broadcasts -> e_new, BN partials
  k_edge<<<dim3(NV, H_ / 16), 32, 0, stream>>>(
      e, W_C, b_C, ws, enew, ws + WS_PSUM, ws + WS_PSQ);

  // K3: gated aggregations (straight and transposed)
  k_hagg<<<NV, H_, 0, stream>>>(enew, ws, ws + WS_H1PRE, 0);
  k_hagg<<<NV, H_, 0, stream>>>(enew, ws, ws + WS_H2PRE, 1);

  // K4: BN+ReLU+residual for h1/h2
  k_bn_h<<<dim3(H_, 2), 256, 0, stream>>>(ws, h1, h2, g_h1, be_h1, g_h2, be_h2,
                                          out);

  // K5: e-BN stats -> scale/shift
  k_bn_e_stats<<<H_, 256, 0, stream>>>(ws + WS_PSUM, ws + WS_PSQ, g_e, be_e,
                                       ws + WS_ESC, ws + WS_ESH);

  // K6: fused BN+ReLU+residual on e, in place
  k_e_final<<<EDGE_F / 4 / 256, 256, 0, stream>>>(enew, e, ws + WS_ESC,
                                                  ws + WS_ESH);
}